// GammatoneFilterbank_56092272886025
// MI455X (gfx1250) — compile-verified
//
#include <hip/hip_runtime.h>

// ---------------------------------------------------------------------------
// Gammatone APGF filterbank on MI455X (gfx1250).
//
// Strategy: exact block state-space reformulation of the IIR cascade.
// For each 2nd-order section  y_t = b0*x_t - a1*y_{t-1} - a2*y_{t-2} :
//   over a 16-sample block with boundary (y_-1, y_-2):
//     y = T * x + p*y_-1 + q*y_-2
//   T = 16x16 lower-triangular Toeplitz of impulse response h,
//   p,q = homogeneous responses (16-vectors).
// A wave processes 256 samples (16 columns x 16) at a time:
//   * zero-state response of all 16 columns = one 16x16 @ 16x16 matmul
//     = 4 chained v_wmma_f32_16x16x4_f32 (A-slices live in VGPRs forever)
//   * 16-step 2x2 boundary scan across columns (readlane + FMA)
//   * rank-2 vector correction  acc += P*EL1 + Q*EL2
// Serial chain: 64000 steps -> 250 macro steps. 512 waves (one per (b,f)).
// ---------------------------------------------------------------------------

typedef __attribute__((ext_vector_type(2))) float v2f;
typedef __attribute__((ext_vector_type(4))) float v4f;
typedef __attribute__((ext_vector_type(8))) float v8f;

#define N_FILTERS 64
#define ORDER     4
#define BATCH     8
#define T_LEN     64000
#define BLK       16                 // samples per column (WMMA M)
#define NCOL      16                 // columns per macro tile (WMMA N)
#define MACRO     (BLK * NCOL)       // 256 samples per macro tile
#define NMACRO    (T_LEN / MACRO)    // 250 macro tiles (exact)

// ---------------------------------------------------------------------------
// Precompute per (filter, section): h[16], p[16], q[16] into d_ws.
// ws layout: coef[f][k][3][16]   (h, p, q)
// ---------------------------------------------------------------------------
__global__ void apgf_precompute(const float* __restrict__ sos,
                                float* __restrict__ coef) {
    int id = threadIdx.x;            // 0..255 -> (f, k)
    int f = id >> 2, k = id & 3;
    float b0 = sos[(f * ORDER + k) * 6 + 0];
    float a1 = sos[(f * ORDER + k) * 6 + 4];
    float a2 = sos[(f * ORDER + k) * 6 + 5];
    float* base = coef + (size_t)((f * ORDER + k) * 3) * BLK;
    // h[-1]=h[-2]=0, driven by b0*delta; p seeded by y_-1=1; q by y_-2=1.
    float hm1 = 0.f, hm2 = 0.f;
    float pm1 = 1.f, pm2 = 0.f;
    float qm1 = 0.f, qm2 = 1.f;
    for (int t = 0; t < BLK; ++t) {
        float h = (t == 0 ? b0 : 0.f) - a1 * hm1 - a2 * hm2;
        float p = -a1 * pm1 - a2 * pm2;
        float q = -a1 * qm1 - a2 * qm2;
        base[t]           = h;
        base[BLK + t]     = p;
        base[2 * BLK + t] = q;
        hm2 = hm1; hm1 = h;
        pm2 = pm1; pm1 = p;
        qm2 = qm1; qm1 = q;
    }
}

__device__ __forceinline__ float rdlane(float v, int lane) {
    return __builtin_bit_cast(float,
        __builtin_amdgcn_readlane(__builtin_bit_cast(int, v), lane));
}

// ---------------------------------------------------------------------------
// Main kernel: one wave per (b, f) channel; 32-thread blocks so barriers are
// single-wave no-ops (they still order DS traffic).
// ---------------------------------------------------------------------------
__global__ void __launch_bounds__(32)
apgf_main(const float* __restrict__ x,
          const float* __restrict__ coef,
          float* __restrict__ out) {
    __shared__ __align__(16) float tile[MACRO];   // 256 floats, t-order

    const int lane   = threadIdx.x;        // 0..31
    const int laneLo = lane & 15;
    const int half   = lane >> 4;          // 0: lanes 0-15, 1: lanes 16-31
    const int chain  = blockIdx.x;         // 0..511
    const int f      = chain & (N_FILTERS - 1);
    const int b      = chain >> 6;

    // ---- per-section constants, resident in VGPRs across all 250 iters ----
    // A-matrix (16x4 f32) layout: lanes 0-15 -> M=lane, vgpr {x,y} = K{0,1};
    //                             lanes 16-31 -> M=lane-16, {x,y} = K{2,3}.
    v2f   A[ORDER][4];
    float Pv[ORDER][8], Qv[ORDER][8];      // C/D-layout aligned p,q vectors
    float p14[ORDER], p15[ORDER], q14[ORDER], q15[ORDER];
    float e1c[ORDER] = {0.f, 0.f, 0.f, 0.f};   // carried y_{-1} per section
    float e2c[ORDER] = {0.f, 0.f, 0.f, 0.f};   // carried y_{-2} per section

#pragma unroll
    for (int k = 0; k < ORDER; ++k) {
        const float* hk = coef + (size_t)((f * ORDER + k) * 3) * BLK;
        const float* pk = hk + BLK;
        const float* qk = hk + 2 * BLK;
#pragma unroll
        for (int s = 0; s < 4; ++s) {
            int j0 = 4 * s + 2 * half;     // global K column of this vgpr
            A[k][s].x = (laneLo >= j0    ) ? hk[laneLo - j0      ] : 0.f;
            A[k][s].y = (laneLo >= j0 + 1) ? hk[laneLo - (j0 + 1)] : 0.f;
        }
        // D-layout: vgpr v holds row M=v (lanes 0-15) / M=v+8 (lanes 16-31)
#pragma unroll
        for (int v = 0; v < 8; ++v) {
            Pv[k][v] = pk[v + 8 * half];
            Qv[k][v] = qk[v + 8 * half];
        }
        p14[k] = pk[14]; p15[k] = pk[15];
        q14[k] = qk[14]; q15[k] = qk[15];
    }

    const float* xrow = x + (size_t)b * T_LEN;
    float*       orow = out + ((size_t)b * N_FILTERS + f) * (size_t)T_LEN;

    for (int mb = 0; mb < NMACRO; ++mb) {
        const int t0 = mb * MACRO;

        // ---- stage 256 input samples into LDS, t-order (2x b128/lane) ----
        {
            const v4f* src  = (const v4f*)(xrow + t0);
            v4f*       dst4 = (v4f*)tile;
            dst4[lane * 2 + 0] = src[lane * 2 + 0];
            dst4[lane * 2 + 1] = src[lane * 2 + 1];
            if (mb + 1 < NMACRO)  // pull next macro tile toward the caches
                __builtin_prefetch(xrow + t0 + MACRO, 0, 3);
        }
        __syncthreads();

#pragma unroll
        for (int k = 0; k < ORDER; ++k) {
            // ---- zero-state response: acc = T_k(16x16) @ X(16x16) --------
            // B-slice (4x16 f32): lanes 0-15 -> N=lane, {x,y} = K{0,1};
            //                     lanes 16-31 -> N=lane-16, {x,y} = K{2,3}.
            // X[i][c] = tile[c*16 + i]  ->  consecutive pair = ds_load_b64.
            v8f acc = {0.f, 0.f, 0.f, 0.f, 0.f, 0.f, 0.f, 0.f};
#pragma unroll
            for (int s = 0; s < 4; ++s) {
                int kx = 4 * s + 2 * half;
                v2f bslc = *(const v2f*)&tile[laneLo * 16 + kx];
                acc = __builtin_amdgcn_wmma_f32_16x16x4_f32(
                    false, A[k][s], false, bslc, (short)0, acc, false, false);
            }

            // ---- boundary scan across the 16 columns (2x2 recurrence) ----
            // rows 14,15 of column c live in acc[6],acc[7] at lane 16+c.
            float e1 = e1c[k], e2 = e2c[k];
            float EL1 = 0.f, EL2 = 0.f;
#pragma unroll
            for (int c = 0; c < NCOL; ++c) {
                EL1 = (laneLo == c) ? e1 : EL1;   // entering y_-1 for col c
                EL2 = (laneLo == c) ? e2 : EL2;   // entering y_-2 for col c
                float y15 = rdlane(acc[7], 16 + c) + p15[k] * e1 + q15[k] * e2;
                float y14 = rdlane(acc[6], 16 + c) + p14[k] * e1 + q14[k] * e2;
                e1 = y15; e2 = y14;
            }
            e1c[k] = e1; e2c[k] = e2;             // carry to next macro tile

            // ---- rank-2 correction, fully vectorized --------------------
#pragma unroll
            for (int v = 0; v < 8; ++v)
                acc[v] += Pv[k][v] * EL1 + Qv[k][v] * EL2;

            // ---- corrected section output -> LDS t-order (next section) -
            __syncthreads();   // all B-slice reads done before overwrite
            {
                float* p  = &tile[laneLo * 16 + 8 * half];
                v4f lo = { acc[0], acc[1], acc[2], acc[3] };
                v4f hi = { acc[4], acc[5], acc[6], acc[7] };
                *((v4f*)p)     = lo;
                *((v4f*)p + 1) = hi;
            }
            __syncthreads();
        }

        // ---- store final 256 samples, coalesced b128 --------------------
        {
            const v4f* t4   = (const v4f*)tile;
            v4f*       dsto = (v4f*)(orow + t0);
            dsto[lane * 2 + 0] = t4[lane * 2 + 0];
            dsto[lane * 2 + 1] = t4[lane * 2 + 1];
        }
        __syncthreads();   // tile reuse next iteration
    }
}

// ---------------------------------------------------------------------------
extern "C" void kernel_launch(void* const* d_in, const int* in_sizes, int n_in,
                              void* d_out, int out_size, void* d_ws, size_t ws_size,
                              hipStream_t stream) {
    const float* x    = (const float*)d_in[0];   // (8, 64000) f32
    const float* sos  = (const float*)d_in[1];   // (64, 4, 6) f32
    float*       out  = (float*)d_out;           // (8, 64, 64000) f32
    float*       coef = (float*)d_ws;            // 64*4*3*16 floats = 48 KB

    apgf_precompute<<<1, N_FILTERS * ORDER, 0, stream>>>(sos, coef);
    apgf_main<<<BATCH * N_FILTERS, 32, 0, stream>>>(x, coef, out);
}